// Resize_52956946759708
// MI455X (gfx1250) — compile-verified
//
#include <hip/hip_runtime.h>

// CDNA5 / gfx1250, wave32. V_WMMA_F32_16X16X4_F32 (full fp32 precision,
// 2048 FLOP/instr) for the three fused spline contractions. Each wave now
// computes a 16x32 output slab (2 accumulators) so the A-operand load stream
// is shared by two WMMAs -> 2x FLOP per byte through the load pipe.

typedef __attribute__((ext_vector_type(2))) float v2f;
typedef __attribute__((ext_vector_type(8))) float v8f;

#define N_IN  96
#define N_OUT 192

__device__ __forceinline__ v8f wmma_f32(v2f a, v2f b, v8f c) {
    return __builtin_amdgcn_wmma_f32_16x16x4_f32(false, a, false, b,
                                                 (short)0, c, false, false);
}

// ---------------------------------------------------------------------------
// Build M = B * inv(A) in fp64 on device. A is tridiagonal: diag 2/3 (5/6 at
// both ends due to dct2 fold), off-diag 1/6. Solve A y = B[o]^T per row o via
// Thomas algorithm; 192 independent systems, one thread each.
// ---------------------------------------------------------------------------
__global__ __launch_bounds__(192) void build_M_kernel(float* __restrict__ M,
                                                      double* __restrict__ scratch) {
    __shared__ double cp[N_IN];    // Thomas superdiag multipliers (shared by all rows)
    __shared__ double minv[N_IN];  // 1 / pivot
    const double e = 1.0 / 6.0;
    const int t = threadIdx.x;     // output row o = t, 0..191

    if (t == 0) {
        double d0 = 5.0 / 6.0;
        minv[0] = 1.0 / d0;
        cp[0]   = e * minv[0];
        for (int i = 1; i < N_IN; ++i) {
            double di = (i == N_IN - 1) ? 5.0 / 6.0 : 2.0 / 3.0;
            double m  = di - e * cp[i - 1];
            minv[i] = 1.0 / m;
            cp[i]   = e * minv[i];
        }
    }
    __syncthreads();

    double* rhs = scratch + (size_t)t * N_IN;
    for (int i = 0; i < N_IN; ++i) rhs[i] = 0.0;

    // Row o of resize matrix B: cubic B-spline weights at taps b-1..b+2, dct2 fold.
    double x  = (double)t * (double)(N_IN - 1) / (double)(N_OUT - 1);
    double fb = floor(x);
    double tt = x - fb;
    int    bi = (int)fb;
    double w[4];
    w[0] = (1.0 - tt) * (1.0 - tt) * (1.0 - tt) / 6.0;
    w[1] = (3.0 * tt * tt * tt - 6.0 * tt * tt + 4.0) / 6.0;
    w[2] = (-3.0 * tt * tt * tt + 3.0 * tt * tt + 3.0 * tt + 1.0) / 6.0;
    w[3] = tt * tt * tt / 6.0;
    for (int k = 0; k < 4; ++k) {
        int idx = bi - 1 + k;
        int m2  = idx % (2 * N_IN); if (m2 < 0) m2 += 2 * N_IN;
        int f   = (m2 >= N_IN) ? (2 * N_IN - 1 - m2) : m2;
        rhs[f] += w[k];
    }

    // Thomas solve A y = rhs (in place).
    rhs[0] = rhs[0] * minv[0];
    for (int i = 1; i < N_IN; ++i)
        rhs[i] = (rhs[i] - e * rhs[i - 1]) * minv[i];
    for (int i = N_IN - 2; i >= 0; --i)
        rhs[i] = rhs[i] - cp[i] * rhs[i + 1];

    for (int i = 0; i < N_IN; ++i)
        M[t * N_IN + i] = (float)rhs[i];
}

// ---------------------------------------------------------------------------
// Pass over the LAST (contiguous) axis: out[p, o] = sum_i in[p, i] * M[o, i].
// One wave per 16x32 slab: A tile (16 p-rows) loaded once, shared by two
// B tiles (o-cols o0..o0+15 and o0+16..o0+31). K=96 in steps of 4.
// ---------------------------------------------------------------------------
__global__ __launch_bounds__(128) void pass_last_kernel(const float* __restrict__ in,
                                                        const float* __restrict__ M,
                                                        float* __restrict__ out) {
    const int wave = (int)((blockIdx.x * blockDim.x + threadIdx.x) >> 5);
    const int lane = threadIdx.x & 31;
    const int opair = wave % (N_OUT / 32);   // 6 slabs of 32 o-cols
    const int ptile = wave / (N_OUT / 32);
    const int r0 = ptile * 16;
    const int o0 = opair * 32;
    const int half = lane >> 4;   // K pair select: 0 -> K{0,1}, 1 -> K{2,3}
    const int l    = lane & 15;

    const float* arow  = in + (size_t)(r0 + l) * N_IN + 2 * half;
    const float* brow0 = M  + (size_t)(o0 + l) * N_IN + 2 * half;
    const float* brow1 = brow0 + 16 * N_IN;

    v8f c0 = {}, c1 = {};
    for (int k = 0; k < N_IN; k += 4) {
        v2f a  = *(const v2f*)(arow + k);
        v2f b0 = *(const v2f*)(brow0 + k);
        v2f b1 = *(const v2f*)(brow1 + k);
        c0 = wmma_f32(a, b0, c0);
        c1 = wmma_f32(a, b1, c1);
    }

    float* orow = out + (size_t)r0 * N_OUT + o0;
#pragma unroll
    for (int v = 0; v < 8; ++v) {
        int m = v + 8 * half;                    // D row (p)
        orow[(size_t)m * N_OUT + l]      = c0[v];
        orow[(size_t)m * N_OUT + 16 + l] = c1[v];
    }
}

// ---------------------------------------------------------------------------
// Pass over a STRIDED axis: out[p, o, q] = sum_i M[o, i] * in[p, i, q].
// One wave per 16x32 slab: A tile = M rows (16 o) loaded once, shared by two
// B tiles (q-cols q0..q0+15 and q0+16..q0+31, stride Q between K rows).
// ---------------------------------------------------------------------------
__global__ __launch_bounds__(128) void pass_mid_kernel(const float* __restrict__ in,
                                                       const float* __restrict__ M,
                                                       float* __restrict__ out,
                                                       int Q, int Qslabs) {
    const int wave = (int)((blockIdx.x * blockDim.x + threadIdx.x) >> 5);
    const int lane = threadIdx.x & 31;
    const int qslab = wave % Qslabs;
    const int rest  = wave / Qslabs;
    const int otile = rest % (N_OUT / 16);
    const int p     = rest / (N_OUT / 16);
    const int o0 = otile * 16;
    const int q0 = qslab * 32;
    const int half = lane >> 4;
    const int l    = lane & 15;

    const float* arow  = M  + (size_t)(o0 + l) * N_IN + 2 * half;
    const float* bbase = in + (size_t)p * N_IN * Q + (size_t)(2 * half) * Q + q0 + l;

    v8f c0 = {}, c1 = {};
    for (int k = 0; k < N_IN; k += 4) {
        v2f a = *(const v2f*)(arow + k);
        v2f b0, b1;
        const float* bk = bbase + (size_t)k * Q;
        b0.x = bk[0];      b0.y = bk[Q];
        b1.x = bk[16];     b1.y = bk[Q + 16];
        c0 = wmma_f32(a, b0, c0);
        c1 = wmma_f32(a, b1, c1);
    }

    float* obase = out + (size_t)p * N_OUT * Q + (size_t)o0 * Q + q0 + l;
#pragma unroll
    for (int v = 0; v < 8; ++v) {
        int m = v + 8 * half;                 // D row (o)
        obase[(size_t)m * Q]      = c0[v];    // q-cols lane&15 and +16
        obase[(size_t)m * Q + 16] = c1[v];
    }
}

// ---------------------------------------------------------------------------
extern "C" void kernel_launch(void* const* d_in, const int* in_sizes, int n_in,
                              void* d_out, int out_size, void* d_ws, size_t ws_size,
                              hipStream_t stream) {
    (void)in_sizes; (void)n_in; (void)out_size; (void)ws_size;

    const float* img = (const float*)d_in[0];      // (2,2,96,96,96) fp32
    float*       out = (float*)d_out;              // (2,2,192,192,192) fp32

    // Workspace layout (needs ~85.2 MB):
    //   [0, 73728)        : M (192x96 fp32)
    //   [+147456)         : fp64 Thomas scratch (192x96 doubles)
    //   Y1 : 36864*192 fp32  (28.3 MB)
    //   Y2 : 384*192*192 fp32 (56.6 MB)
    char*   ws      = (char*)d_ws;
    float*  M       = (float*)ws;
    double* scratch = (double*)(ws + 73728);
    float*  Y1      = (float*)(ws + 73728 + 147456);
    float*  Y2      = (float*)(ws + 73728 + 147456 + (size_t)36864 * 192 * 4);

    // 1) Fused matrix M = B * inv(A), fp64 Thomas solve.
    build_M_kernel<<<1, 192, 0, stream>>>(M, scratch);

    // 2) Contract last axis (x): (36864, 96) -> (36864, 192).
    {
        int waves = (36864 / 16) * (N_OUT / 32);   // 13824 slab-waves
        pass_last_kernel<<<waves * 32 / 128, 128, 0, stream>>>(img, M, Y1);
    }
    // 3) Contract y axis: (384, 96, 192) -> (384, 192, 192).
    {
        int Q = 192, Qslabs = Q / 32;              // 6
        int waves = 384 * (N_OUT / 16) * Qslabs;   // 27648 slab-waves
        pass_mid_kernel<<<waves * 32 / 128, 128, 0, stream>>>(Y1, M, Y2, Q, Qslabs);
    }
    // 4) Contract z axis: (4, 96, 36864) -> (4, 192, 36864) = final output.
    {
        int Q = 192 * 192, Qslabs = Q / 32;        // 1152
        int waves = 4 * (N_OUT / 16) * Qslabs;     // 55296 slab-waves
        pass_mid_kernel<<<waves * 32 / 128, 128, 0, stream>>>(Y2, M, out, Q, Qslabs);
    }
}